// EvolveGCN_24489903522226
// MI455X (gfx1250) — compile-verified
//
#include <hip/hip_runtime.h>
#include <math.h>
#include <stdint.h>

// Problem sizes (fixed by the reference).
#define Nn     60000
#define Ee     200000
#define Dd     128
#define NINPc  256
#define Tt     6
#define Ll     2
#define Bb     16
#define GG     (4*Dd)   // 512 gate columns

typedef float v2f __attribute__((ext_vector_type(2)));
typedef float v8f __attribute__((ext_vector_type(8)));

// D(16x16 f32) = A(16x4 f32) * B(4x16 f32) + C   — CDNA5 WMMA, wave32.
__device__ __forceinline__ v8f wmma4(v2f a, v2f b, v8f c) {
  return __builtin_amdgcn_wmma_f32_16x16x4_f32(
      /*neg_a=*/false, a, /*neg_b=*/false, b,
      /*c_mod=*/(short)0, c, /*reuse_a=*/false, /*reuse_b=*/false);
}

// ---- CDNA5 async global->LDS copies (ASYNCcnt-tracked) ----
__device__ __forceinline__ unsigned lds_off(const void* p) {
  // flat addr of LDS: low 32 bits are the wave-relative LDS byte offset.
  return (unsigned)(uintptr_t)p;
}
__device__ __forceinline__ void async_b128(unsigned lds, unsigned long long g) {
  asm volatile("global_load_async_to_lds_b128 %0, %1, off" :: "v"(lds), "v"(g) : "memory");
}
__device__ __forceinline__ void async_b64(unsigned lds, unsigned long long g) {
  asm volatile("global_load_async_to_lds_b64 %0, %1, off" :: "v"(lds), "v"(g) : "memory");
}
template<int N> __device__ __forceinline__ void wait_async() {
  asm volatile("s_wait_asynccnt %0" :: "n"(N) : "memory");
}

// =====================================================================
// NN GEMM: C[M x NB] = A[M x K] (optionally row-gathered) * B[K x NB] (+bias)
// Async double-buffered K=32 tiles staged via GLOBAL_LOAD_ASYNC_TO_LDS.
// Block = 256 threads (8 waves); tile = 16 rows x 128 cols.
// Requires M % 16 == 0, K % 32 == 0 (true for all uses here).
// =====================================================================
template<bool GATHER_A>
__global__ __launch_bounds__(256) void gemm_nn_async(
    const float* __restrict__ A, const int* __restrict__ rowids,
    const float* __restrict__ Bm, const float* __restrict__ bias,
    float* __restrict__ C, int M, int K, int NB) {
  __shared__ float As[2][16][36];    // pad 36: 4r mod 64 distinct -> conflict-free
  __shared__ float Bs[2][32][136];   // pad 136: 2*136 mod 64 = 16 -> half-waves disjoint banks

  const int tid  = threadIdx.x;
  const int wave = tid >> 5;
  const int lane = tid & 31;
  const int m0   = blockIdx.x * 16;
  const int by   = blockIdx.y;
  const int mrow = lane & 15;
  const int khalf = (lane >> 4) << 1;      // lanes 0-15 -> K {0,1}; 16-31 -> K {2,3}

  // Fixed per-thread staging coordinates.
  const int ar_r = tid >> 4;               // A row 0..15
  const int ar_c = (tid & 15) * 2;         // A col 0..30 (b64 = 2 floats)
  int arow = m0 + ar_r;
  if (GATHER_A) arow = rowids[arow];
  const float* aptr = A + (size_t)arow * K + ar_c;

  // Stage one 32-wide K tile into buffer s. Per wave: 1 b64 + 4 b128 = 5 async ops.
  auto stage = [&](int kt, int s) {
    async_b64(lds_off(&As[s][ar_r][ar_c]),
              (unsigned long long)(uintptr_t)(aptr + kt));
    #pragma unroll
    for (int it = 0; it < 4; ++it) {
      const float* g = Bm + (size_t)(kt + wave + it * 8) * NB + by * 128 + lane * 4;
      async_b128(lds_off(&Bs[s][wave + it * 8][lane * 4]),
                 (unsigned long long)(uintptr_t)g);
    }
  };

  v8f acc = {};
  int buf = 0;
  stage(0, 0);
  const int NKT = K >> 5;
  for (int ki = 0; ki < NKT; ++ki) {
    if (ki + 1 < NKT) {                    // prefetch next tile into other buffer
      stage((ki + 1) << 5, buf ^ 1);
      wait_async<5>();                     // oldest 5 (current tile) have landed
    } else {
      wait_async<0>();
    }
    __syncthreads();
    #pragma unroll
    for (int kk = 0; kk < 32; kk += 4) {
      v2f a, b;
      a.x = As[buf][mrow][kk + khalf + 0];
      a.y = As[buf][mrow][kk + khalf + 1];
      b.x = Bs[buf][kk + khalf + 0][wave * 16 + mrow];
      b.y = Bs[buf][kk + khalf + 1][wave * 16 + mrow];
      acc = wmma4(a, b, acc);
    }
    __syncthreads();                       // protect buffer reuse next iteration
    buf ^= 1;
  }

  // C/D layout: VGPR r, lane l -> row m0 + r + (l>=16 ? 8 : 0), col = lane%16.
  const int colg = by * 128 + wave * 16 + mrow;
  const float bv = bias ? bias[colg] : 0.0f;
  #pragma unroll
  for (int r = 0; r < 8; ++r) {
    int mr = m0 + r + ((lane >> 4) << 3);
    C[(size_t)mr * NB + colg] = acc[r] + bv;
  }
}

// =====================================================================
// NT GEMM (x @ W.T): C[M x NB] = A[M x K] * Bm[NB x K]^T — used for the tiny
// LSTM gate GEMMs; synchronous staging (transpose needs 4B-scattered LDS writes).
// =====================================================================
#define KTILE 64
__global__ __launch_bounds__(256) void gemm_nt(
    const float* __restrict__ A, const float* __restrict__ Bm,
    float* __restrict__ C, int M, int K, int NB) {
  __shared__ float As[16][KTILE + 4];
  __shared__ float Bs[KTILE][128 + 4];

  const int tid  = threadIdx.x;
  const int wave = tid >> 5;
  const int lane = tid & 31;
  const int m0   = blockIdx.x * 16;
  const int mrow = lane & 15;
  const int khalf = (lane >> 4) << 1;

  v8f acc = {};
  for (int kt = 0; kt < K; kt += KTILE) {
    {
      int idx = tid * 4;
      int r = idx / KTILE, c = idx % KTILE;
      float4 v = *(const float4*)(A + (size_t)(m0 + r) * K + kt + c);
      As[r][c+0]=v.x; As[r][c+1]=v.y; As[r][c+2]=v.z; As[r][c+3]=v.w;
    }
    #pragma unroll
    for (int it = 0; it < 8; ++it) {
      int idx = (tid + it * 256) * 4;
      int nn = idx / KTILE, kk = idx % KTILE;   // coalesced along K of Bm[NB x K]
      float4 v = *(const float4*)(Bm + (size_t)(blockIdx.y * 128 + nn) * K + kt + kk);
      Bs[kk+0][nn]=v.x; Bs[kk+1][nn]=v.y; Bs[kk+2][nn]=v.z; Bs[kk+3][nn]=v.w;
    }
    __syncthreads();
    #pragma unroll
    for (int kk = 0; kk < KTILE; kk += 4) {
      v2f a, b;
      a.x = As[mrow][kk + khalf + 0];
      a.y = As[mrow][kk + khalf + 1];
      b.x = Bs[kk + khalf + 0][wave * 16 + mrow];
      b.y = Bs[kk + khalf + 1][wave * 16 + mrow];
      acc = wmma4(a, b, acc);
    }
    __syncthreads();
  }
  const int colg = blockIdx.y * 128 + wave * 16 + mrow;
  #pragma unroll
  for (int r = 0; r < 8; ++r) {
    int mr = m0 + r + ((lane >> 4) << 3);
    C[(size_t)mr * NB + colg] = acc[r];
  }
}

// ---------------- utility kernels ----------------
__global__ void fill_f32(float* p, float v, int n) {
  int i = blockIdx.x * blockDim.x + threadIdx.x; if (i < n) p[i] = v;
}
__global__ void fill_i32(int* p, int v, int n) {
  int i = blockIdx.x * blockDim.x + threadIdx.x; if (i < n) p[i] = v;
}
__global__ void fill_u32(unsigned* p, unsigned v, int n) {
  int i = blockIdx.x * blockDim.x + threadIdx.x; if (i < n) p[i] = v;
}
__global__ void copy_f32(float* d, const float* s, int n) {
  int i = blockIdx.x * blockDim.x + threadIdx.x; if (i < n) d[i] = s[i];
}

__global__ void compute_has(const int* __restrict__ et, int* has, int E) {
  int e = blockIdx.x * blockDim.x + threadIdx.x;
  if (e < E) atomicOr(&has[et[e]], 1);
}
__global__ void compute_first(const int* has, int* firstb, int Tn) {
  if (blockIdx.x == 0 && threadIdx.x == 0) {
    int a = 0;
    for (int t = 0; t < Tn; ++t) { firstb[t] = a; a |= has[t]; }
  }
}
__global__ void mark_touched(const int* __restrict__ et, const int* __restrict__ src,
                             const int* __restrict__ dst, int t, int* touched, int E) {
  int e = blockIdx.x * blockDim.x + threadIdx.x;
  if (e >= E) return;
  if (et[e] == t) { touched[src[e]] = 1; touched[dst[e]] = 1; }
}

__device__ __forceinline__ float sigm(float x) { return 1.0f / (1.0f + expf(-x)); }

// LSTM weight-evolution elementwise update (per layer).
__global__ void lstm_update(const float* __restrict__ M1, const float* __restrict__ M2,
                            const float* __restrict__ bih, const float* __restrict__ bhh,
                            float* __restrict__ hx, float* __restrict__ cx,
                            const int* has, const int* firstb, int t) {
  int idx = blockIdx.x * blockDim.x + threadIdx.x;
  if (idx >= Dd * Dd) return;
  int m = idx / Dd, d = idx % Dd;
  bool hasv  = has[t] != 0;
  bool first = firstb[t] != 0;

  float i0 = M1[(size_t)m*GG +        d] + bih[       d] + bhh[       d];
  float f0 = M1[(size_t)m*GG +   Dd + d] + bih[  Dd + d] + bhh[  Dd + d];
  float g0 = M1[(size_t)m*GG + 2*Dd + d] + bih[2*Dd + d] + bhh[2*Dd + d];
  float o0 = M1[(size_t)m*GG + 3*Dd + d] + bih[3*Dd + d] + bhh[3*Dd + d];
  float is = i0 + M2[(size_t)m*GG +        d];
  float fs = f0 + M2[(size_t)m*GG +   Dd + d];
  float gs = g0 + M2[(size_t)m*GG + 2*Dd + d];
  float os = o0 + M2[(size_t)m*GG + 3*Dd + d];

  float cxv = cx[idx];
  float c2z = sigm(i0) * tanhf(g0);           // zero-state cell (f-gate term drops)
  float hz  = sigm(o0) * tanhf(c2z);
  float c2s = sigm(fs) * cxv + sigm(is) * tanhf(gs);
  float hs  = sigm(os) * tanhf(c2s);

  float hi = first ? hs  : hz;
  float ci = first ? cxv : c2z;               // faithful "frozen cx" semantics
  if (!hasv) { hi = hx[idx]; ci = cxv; }
  hx[idx] = hi;
  cx[idx] = ci;
}

// One wave per edge: agg[dst] += hw[src] * w for active edges.
__global__ void edge_scatter(const int* __restrict__ et, const int* __restrict__ src,
                             const int* __restrict__ dst, const float* __restrict__ ew,
                             const float* __restrict__ hw, float* __restrict__ agg, int t) {
  int gid = blockIdx.x * blockDim.x + threadIdx.x;
  int e = gid >> 5, lane = gid & 31;
  if (e >= Ee) return;
  if (et[e] != t) return;
  float w = ew[e];
  const float* srow = hw + (size_t)src[e] * Dd;
  float*       drow = agg + (size_t)dst[e] * Dd;
  __builtin_prefetch(srow, 0, 1);   // global_prefetch_b8
  #pragma unroll
  for (int k = lane; k < Dd; k += 32) atomicAdd(&drow[k], srow[k] * w);
}

__global__ void relu_inplace(float* __restrict__ x, int count4) {
  int i = blockIdx.x * blockDim.x + threadIdx.x;
  if (i >= count4) return;
  float4* p = (float4*)x + i;
  float4 v = *p;
  v.x = fmaxf(v.x, 0.f); v.y = fmaxf(v.y, 0.f);
  v.z = fmaxf(v.z, 0.f); v.w = fmaxf(v.w, 0.f);
  *p = v;
}

__global__ void writeback(float* __restrict__ h, const float* __restrict__ ht,
                          const int* __restrict__ touched) {
  int idx = blockIdx.x * blockDim.x + threadIdx.x;   // N*32 threads, float4 each
  int n = idx >> 5, c = (idx & 31) * 4;
  if (n >= Nn) return;
  if (touched[n]) {
    float4 v = *(const float4*)(ht + (size_t)n * Dd + c);
    *(float4*)(h + (size_t)n * Dd + c) = v;
  }
}

// Ordered-uint encoding so unsigned atomicMax == float max.
__device__ __forceinline__ unsigned fenc(float f) {
  unsigned u = __float_as_uint(f);
  return (u & 0x80000000u) ? ~u : (u | 0x80000000u);
}
__device__ __forceinline__ float fdec(unsigned u) {
  return __uint_as_float((u & 0x80000000u) ? (u & 0x7FFFFFFFu) : ~u);
}

// Segment max over sorted graph_id; block = 128 cols x 64-node strip.
__global__ void pool_max(const float* __restrict__ h, const int* __restrict__ gid,
                         unsigned* __restrict__ pooled) {
  int d = threadIdx.x;
  int nbase = blockIdx.x * 64;
  int cur = -1; float best = -INFINITY;
  for (int j = 0; j < 64; ++j) {
    int n = nbase + j;
    if (n >= Nn) break;
    int g = gid[n];
    if (g != cur) {
      if (cur >= 0) atomicMax(&pooled[(size_t)cur * Dd + d], fenc(best));
      cur = g; best = -INFINITY;
    }
    best = fmaxf(best, h[(size_t)n * Dd + d]);
  }
  if (cur >= 0) atomicMax(&pooled[(size_t)cur * Dd + d], fenc(best));
}

// Head: logits = pooled @ out_W + b; BCE-with-logits mean; sigmoid probs.
__global__ void final_head(const unsigned* __restrict__ pooled, const float* __restrict__ outW,
                           const float* __restrict__ outb, const float* __restrict__ y,
                           float* __restrict__ out) {
  __shared__ float logits[Bb];
  int b = threadIdx.x;
  if (b < Bb) {
    float acc = 0.f;
    for (int dd = 0; dd < Dd; ++dd) {
      float p = fdec(pooled[(size_t)b * Dd + dd]);
      if (!isfinite(p)) p = 0.f;
      acc += p * outW[dd];
    }
    float lg = acc + outb[0];
    logits[b] = lg;
    out[1 + b] = 1.f / (1.f + expf(-lg));
  }
  __syncthreads();
  if (threadIdx.x == 0) {
    float s = 0.f;
    for (int i = 0; i < Bb; ++i) {
      float lg = logits[i];
      s += fmaxf(lg, 0.f) - lg * y[i] + log1pf(expf(-fabsf(lg)));
    }
    out[0] = s / (float)Bb;
  }
}

extern "C" void kernel_launch(void* const* d_in, const int* in_sizes, int n_in,
                              void* d_out, int out_size, void* d_ws, size_t ws_size,
                              hipStream_t stream) {
  const int*   word_ids    = (const int*)  d_in[0];
  const int*   srcp        = (const int*)  d_in[1];
  const int*   dstp        = (const int*)  d_in[2];
  const int*   edge_time   = (const int*)  d_in[3];
  const float* edge_weight = (const float*)d_in[4];
  const int*   graph_id    = (const int*)  d_in[5];
  const float* y_data      = (const float*)d_in[6];
  const float* word_embeds = (const float*)d_in[7];
  const float* adapt_W     = (const float*)d_in[8];
  const float* adapt_b     = (const float*)d_in[9];
  const float* gcn_w       = (const float*)d_in[10];
  const float* Wih         = (const float*)d_in[11];
  const float* Whh         = (const float*)d_in[12];
  const float* bih         = (const float*)d_in[13];
  const float* bhh         = (const float*)d_in[14];
  const float* outW        = (const float*)d_in[15];
  const float* outb        = (const float*)d_in[16];

  // Workspace carve-up (256B aligned).
  char* base = (char*)d_ws; size_t off = 0;
  auto alloc = [&](size_t bytes) -> void* {
    void* p = base + off; off = (off + bytes + 255) & ~(size_t)255; return p;
  };
  float*    h       = (float*)   alloc((size_t)Nn * Dd * 4);
  float*    hw      = (float*)   alloc((size_t)Nn * Dd * 4);
  float*    agg     = (float*)   alloc((size_t)Nn * Dd * 4);  // doubles as ht (ReLU in place)
  float*    hx      = (float*)   alloc((size_t)Ll * Dd * Dd * 4);
  float*    cx      = (float*)   alloc((size_t)Ll * Dd * Dd * 4);
  float*    M1      = (float*)   alloc((size_t)Dd * GG * 4);
  float*    M2      = (float*)   alloc((size_t)Dd * GG * 4);
  int*      touched = (int*)     alloc((size_t)Nn * 4);
  int*      has     = (int*)     alloc(Tt * 4);
  int*      firstb  = (int*)     alloc(Tt * 4);
  unsigned* pooled  = (unsigned*)alloc((size_t)Bb * Dd * 4);
  (void)ws_size; (void)in_sizes; (void)n_in; (void)out_size;

  // ---- deterministic per-call init ----
  fill_f32<<<(Ll*Dd*Dd + 255)/256, 256, 0, stream>>>(cx, 0.f, Ll*Dd*Dd);
  copy_f32<<<(Ll*Dd*Dd + 255)/256, 256, 0, stream>>>(hx, gcn_w, Ll*Dd*Dd);
  fill_u32<<<(Bb*Dd + 255)/256, 256, 0, stream>>>(pooled, 0x007FFFFFu, Bb*Dd); // fenc(-inf)
  fill_i32<<<1, 32, 0, stream>>>(has, 0, Tt);
  compute_has<<<(Ee + 255)/256, 256, 0, stream>>>(edge_time, has, Ee);
  compute_first<<<1, 1, 0, stream>>>(has, firstb, Tt);

  // h0 = word_embeds[word_ids] @ adapt_W + adapt_b
  gemm_nn_async<true><<<dim3(Nn/16, 1), 256, 0, stream>>>(
      word_embeds, word_ids, adapt_W, adapt_b, h, Nn, NINPc, Dd);

  for (int t = 0; t < Tt; ++t) {
    fill_i32<<<(Nn + 255)/256, 256, 0, stream>>>(touched, 0, Nn);
    mark_touched<<<(Ee + 255)/256, 256, 0, stream>>>(edge_time, srcp, dstp, t, touched, Ee);

    for (int i = 0; i < Ll; ++i) {
      const float* hx_i = hx + (size_t)i * Dd * Dd;
      // LSTM gates: M1 = hx @ Wih.T, M2 = hx @ Whh.T  (W is [512,128])
      gemm_nt<<<dim3(Dd/16, GG/128), 256, 0, stream>>>(
          hx_i, Wih + (size_t)i*GG*Dd, M1, Dd, Dd, GG);
      gemm_nt<<<dim3(Dd/16, GG/128), 256, 0, stream>>>(
          hx_i, Whh + (size_t)i*GG*Dd, M2, Dd, Dd, GG);
      lstm_update<<<(Dd*Dd + 255)/256, 256, 0, stream>>>(
          M1, M2, bih + (size_t)i*GG, bhh + (size_t)i*GG,
          hx + (size_t)i*Dd*Dd, cx + (size_t)i*Dd*Dd, has, firstb, t);

      // hw = ht @ hi  (ht = h for layer 0, previous ReLU output otherwise)
      const float* ht_in = (i == 0) ? h : agg;
      gemm_nn_async<false><<<dim3(Nn/16, 1), 256, 0, stream>>>(
          ht_in, nullptr, hx + (size_t)i*Dd*Dd, nullptr, hw, Nn, Dd, Dd);

      fill_f32<<<((size_t)Nn*Dd + 255)/256, 256, 0, stream>>>(agg, 0.f, Nn*Dd);
      edge_scatter<<<((size_t)Ee*32 + 255)/256, 256, 0, stream>>>(
          edge_time, srcp, dstp, edge_weight, hw, agg, t);
      relu_inplace<<<((size_t)Nn*Dd/4 + 255)/256, 256, 0, stream>>>(agg, Nn*Dd/4);
    }
    writeback<<<((size_t)Nn*32 + 255)/256, 256, 0, stream>>>(h, agg, touched);
  }

  pool_max<<<(Nn + 63)/64, 128, 0, stream>>>(h, graph_id, pooled);
  final_head<<<1, 32, 0, stream>>>(pooled, outW, outb, y_data, (float*)d_out);
}